// ForwardWarpDWeight_13572096656074
// MI455X (gfx1250) — compile-verified
//
#include <hip/hip_runtime.h>

// Forward bilinear splat warp + normalize for MI455X (gfx1250).
// Strategy: direct no-return global_atomic_add_f32 into L2-resident
// accumulators (78.6 MB << 192 MB L2); all streaming traffic uses
// non-temporal th: hints so it does not evict the accumulator set.
// 3D launch geometry (W/256, H, B) removes all integer division.

namespace {

constexpr int B = 8, C = 3, H = 384, W = 1280;
constexpr int HW = H * W;  // 491520

// Native clang vector (HIP's float2 class is rejected by the NT builtins).
typedef float vfloat2 __attribute__((ext_vector_type(2)));

__device__ __forceinline__ void atomic_add_f32(float* p, float v) {
  // Relaxed, agent scope, result unused -> non-returning global_atomic_add_f32.
  (void)__hip_atomic_fetch_add(p, v, __ATOMIC_RELAXED, __HIP_MEMORY_SCOPE_AGENT);
}

// One thread per source pixel: 20 atomics (5 channels x 4 corners) when the
// full 2x2 footprint is in-bounds, else nothing (reference adds zeros).
// Launch: block(256), grid(W/256, H, B) -- exact cover, no guards.
__global__ __launch_bounds__(256) void fwdwarp_splat(
    const float* __restrict__ xin,    // [B,C,H,W]
    const float* __restrict__ flow,   // [B,2,H,W]
    const float* __restrict__ depth,  // [B,1,H,W]
    float* __restrict__ out,          // [B,C,H,W]  (xw accumulators, pre-zeroed)
    float* __restrict__ dm)           // [B*HW][2]  {dw accum, mask accum}, pre-zeroed
{
  const int px = blockIdx.x * 256 + threadIdx.x;
  const int py = blockIdx.y;
  const int b  = blockIdx.z;
  const int p  = py * W + px;

  // Streaming reads: non-temporal so they do not evict L2 accumulators.
  const size_t fbase = (size_t)b * 2 * HW + (size_t)p;
  float fx = __builtin_nontemporal_load(flow + fbase);
  float fy = __builtin_nontemporal_load(flow + fbase + HW);
  fx = fminf(fmaxf(fx, -2.0f * (float)W), 2.0f * (float)W);
  fy = fminf(fmaxf(fy, -2.0f * (float)W), 2.0f * (float)W);

  const float xs = fx + (float)px;
  const float ys = fy + (float)py;
  const float x0 = floorf(xs);
  const float y0 = floorf(ys);
  // valid iff x0>=0 && x0+1<=W-1 && y0>=0 && y0+1<=H-1
  if (!(x0 >= 0.0f && x0 <= (float)(W - 2) &&
        y0 >= 0.0f && y0 <= (float)(H - 2)))
    return;

  float d = __builtin_nontemporal_load(depth + (size_t)b * HW + (size_t)p);
  d = fminf(fmaxf(d, 0.001f), 80.0f);
  const float dw = expf((40.0f - d) * 0.2f);

  const size_t xbase = (size_t)b * C * HW + (size_t)p;
  const float c0 = __builtin_nontemporal_load(xin + xbase) * dw;
  const float c1 = __builtin_nontemporal_load(xin + xbase + (size_t)HW) * dw;
  const float c2 = __builtin_nontemporal_load(xin + xbase + 2 * (size_t)HW) * dw;

  const float ax = xs - x0;
  const float ay = ys - y0;
  const int   q  = (int)y0 * W + (int)x0;

  const int   qs[4] = {q, q + 1, q + W, q + W + 1};
  const float wv[4] = {(1.0f - ax) * (1.0f - ay), ax * (1.0f - ay),
                       (1.0f - ax) * ay,          ax * ay};

  float* const outb = out + (size_t)b * C * HW;
  float* const dmb  = dm + (size_t)b * HW * 2;
#pragma unroll
  for (int k = 0; k < 4; ++k) {
    const float w = wv[k];
    float* dmp = dmb + (size_t)qs[k] * 2;     // dw & mask: adjacent 8 bytes
    atomic_add_f32(dmp + 0, dw * w);
    atomic_add_f32(dmp + 1, w);
    float* ob = outb + (size_t)qs[k];
    atomic_add_f32(ob,                   c0 * w);
    atomic_add_f32(ob + (size_t)HW,      c1 * w);
    atomic_add_f32(ob + 2 * (size_t)HW,  c2 * w);
  }
}

// out = (mask < 0.5 ? 0 : xw) / max(dw, 1e-7), in place on the accumulators.
// Launch: block(256), grid(W/256, H, B).
__global__ __launch_bounds__(256) void fwdwarp_normalize(
    const vfloat2* __restrict__ dm2, float* __restrict__ out)
{
  const int px = blockIdx.x * 256 + threadIdx.x;
  const int py = blockIdx.y;
  const int b  = blockIdx.z;
  const int p  = py * W + px;

  // dw/mask pair is dead after this read -> single non-temporal b64 load.
  const vfloat2 dmv = __builtin_nontemporal_load(dm2 + (size_t)b * HW + (size_t)p);
  const float denom   = fmaxf(dmv.x, 1e-7f);
  const bool  invalid = dmv.y < 0.5f;

  float* ob = out + (size_t)b * C * HW + (size_t)p;
#pragma unroll
  for (int c = 0; c < C; ++c) {
    float v = ob[(size_t)c * HW];             // accumulator: expect L2 hit
    v = invalid ? 0.0f : v / denom;           // IEEE divide to match reference
    __builtin_nontemporal_store(v, ob + (size_t)c * HW);
  }
}

}  // namespace

extern "C" void kernel_launch(void* const* d_in, const int* in_sizes, int n_in,
                              void* d_out, int out_size, void* d_ws, size_t ws_size,
                              hipStream_t stream) {
  (void)in_sizes; (void)n_in; (void)out_size; (void)ws_size;
  const float* xin   = (const float*)d_in[0];  // x     [B,C,H,W]
  const float* flow  = (const float*)d_in[1];  // flow  [B,2,H,W]
  const float* depth = (const float*)d_in[2];  // depth [B,1,H,W]
  float* out = (float*)d_out;                  // [B,C,H,W] f32
  float* dm  = (float*)d_ws;                   // [B*HW][2] f32 = 31.5 MB scratch

  // Zero accumulators every call (harness does not re-poison between replays).
  (void)hipMemsetAsync(out, 0, (size_t)B * C * HW * sizeof(float), stream);
  (void)hipMemsetAsync(dm,  0, (size_t)B * HW * 2 * sizeof(float), stream);

  const dim3 block(256);                 // 8 wave32 waves / block
  const dim3 grid(W / 256, H, B);        // 5 x 384 x 8 = exact pixel cover
  fwdwarp_splat<<<grid, block, 0, stream>>>(xin, flow, depth, out, dm);
  fwdwarp_normalize<<<grid, block, 0, stream>>>((const vfloat2*)dm, out);
}